// GATFeatureAggregator_10445360464273
// MI455X (gfx1250) — compile-verified
//
#include <hip/hip_runtime.h>

// ---------------------------------------------------------------------------
// GAT 2-layer forward for MI455X (gfx1250, wave32).
//   layer1: z1 = x @ W1            [N,64]   (fp32 WMMA 16x16x4)
//           el/er per (node,head), edge softmax over dst, scatter-agg -> h
//           h = elu(agg + b1)
//   layer2: z2 = h @ W2            [N,64]   (fp32 WMMA 16x16x4)
//           single-head edge softmax, scatter-agg -> out, out += b2
// ---------------------------------------------------------------------------

typedef float v2f __attribute__((ext_vector_type(2)));
typedef float v8f __attribute__((ext_vector_type(8)));

#define NEG_SLOPE 0.2f

__device__ __forceinline__ unsigned fenc(float x) {
  unsigned b = __float_as_uint(x);
  return (b & 0x80000000u) ? ~b : (b | 0x80000000u);   // order-preserving map
}
__device__ __forceinline__ float fdec(unsigned k) {
  unsigned b = (k & 0x80000000u) ? (k ^ 0x80000000u) : ~k;
  return __uint_as_float(b);
}
__device__ __forceinline__ float lrelu(float x) { return x > 0.f ? x : NEG_SLOPE * x; }

// ---------------------------------------------------------------------------
__global__ void fill_zero(float* __restrict__ p, long long n) {
  long long i = (long long)blockIdx.x * blockDim.x + threadIdx.x;
  long long stride = (long long)gridDim.x * blockDim.x;
  for (; i < n; i += stride) p[i] = 0.f;
}

// Z[nrows x 64] = X[nrows x K] @ W[K x 64], fp32 WMMA.
// One wave32 per 16-row strip; 4 accumulator tiles cover all 64 columns.
__global__ __launch_bounds__(256)
void gemm_f32_wmma(const float* __restrict__ X, const float* __restrict__ W,
                   float* __restrict__ Z, int nrows, int K) {
  const int Fout = 64;
  int lane = threadIdx.x & 31;
  int wave = threadIdx.x >> 5;
  int rowBlock = blockIdx.x * (blockDim.x >> 5) + wave;
  if (rowBlock * 16 >= nrows) return;            // wave-uniform: EXEC stays all-1s
  int row0 = rowBlock * 16;
  int m    = lane & 15;                          // row within tile (A), col within tile (B/C)
  int koff = (lane >> 4) << 1;                   // 0 for lanes 0-15, 2 for lanes 16-31

  v8f acc0 = {}, acc1 = {}, acc2 = {}, acc3 = {};
  const float* xrow = X + (long long)(row0 + m) * K;

  for (int k0 = 0; k0 < K; k0 += 4) {
    // A 16x4: VGPR0=K(k0+koff), VGPR1=K(k0+koff+1), M = lane&15
    v2f a;
    a.x = xrow[k0 + koff];
    a.y = xrow[k0 + koff + 1];
    // B 4x16 per column tile: VGPR0 row k0+koff, VGPR1 row k0+koff+1, N = tile*16 + m
    const float* w0 = W + (long long)(k0 + koff) * Fout + m;
    const float* w1 = w0 + Fout;
    v2f b0, b1, b2, b3;
    b0.x = w0[0];   b0.y = w1[0];
    b1.x = w0[16];  b1.y = w1[16];
    b2.x = w0[32];  b2.y = w1[32];
    b3.x = w0[48];  b3.y = w1[48];
    acc0 = __builtin_amdgcn_wmma_f32_16x16x4_f32(false, a, false, b0, (short)0, acc0, false, false);
    acc1 = __builtin_amdgcn_wmma_f32_16x16x4_f32(false, a, false, b1, (short)0, acc1, false, false);
    acc2 = __builtin_amdgcn_wmma_f32_16x16x4_f32(false, a, false, b2, (short)0, acc2, false, false);
    acc3 = __builtin_amdgcn_wmma_f32_16x16x4_f32(false, a, false, b3, (short)0, acc3, false, false);
  }

  // C/D layout: VGPR r -> row = row0 + r (+8 for lanes 16-31), col = tile*16 + (lane&15)
  int rbase = row0 + ((lane >> 4) << 3);
  int c = lane & 15;
#pragma unroll
  for (int r = 0; r < 8; ++r) {
    float* zr = Z + (long long)(rbase + r) * Fout + c;
    zr[0]  = acc0[r];
    zr[16] = acc1[r];
    zr[32] = acc2[r];
    zr[48] = acc3[r];
  }
}

// ---------------------------------------------------------------------------
// el/er coefficients, layer 1: 2 heads x 32 feats  (one thread per (node,head))
__global__ void attn_coef_h2(const float* __restrict__ z, const float* __restrict__ al,
                             const float* __restrict__ ar, float* __restrict__ el,
                             float* __restrict__ er, int n_nodes) {
  int gid = blockIdx.x * blockDim.x + threadIdx.x;
  if (gid >= n_nodes * 2) return;
  int n = gid >> 1, h = gid & 1;
  const float* zp  = z + (long long)n * 64 + h * 32;
  const float* alp = al + h * 32;
  const float* arp = ar + h * 32;
  float sl = 0.f, sr = 0.f;
#pragma unroll
  for (int f = 0; f < 32; ++f) { float v = zp[f]; sl += v * alp[f]; sr += v * arp[f]; }
  el[gid] = sl; er[gid] = sr;
}

// el/er coefficients, layer 2: 1 head x 64 feats  (one thread per node)
__global__ void attn_coef_h1(const float* __restrict__ z, const float* __restrict__ al,
                             const float* __restrict__ ar, float* __restrict__ el,
                             float* __restrict__ er, int n_nodes) {
  int gid = blockIdx.x * blockDim.x + threadIdx.x;
  if (gid >= n_nodes) return;
  const float* zp = z + (long long)gid * 64;
  float sl = 0.f, sr = 0.f;
#pragma unroll
  for (int f = 0; f < 64; ++f) { float v = zp[f]; sl += v * al[f]; sr += v * ar[f]; }
  el[gid] = sl; er[gid] = sr;
}

// ---------------------------------------------------------------------------
// Pass 1: per-dst segment max of leaky_relu(el[src]+er[dst]) via uint-key atomicMax
__global__ void edge_max(const int* __restrict__ src, const int* __restrict__ dst,
                         const float* __restrict__ el, const float* __restrict__ er,
                         unsigned* __restrict__ mkey, int E, int H) {
  int gid = blockIdx.x * blockDim.x + threadIdx.x;
  if (gid >= E) return;
  int s = src[gid], d = dst[gid];
  for (int h = 0; h < H; ++h) {
    float e = lrelu(el[s * H + h] + er[d * H + h]);
    atomicMax(&mkey[d * H + h], fenc(e));
  }
}

// Pass 2: per-dst sum of exp(e - max)
__global__ void edge_denom(const int* __restrict__ src, const int* __restrict__ dst,
                           const float* __restrict__ el, const float* __restrict__ er,
                           const unsigned* __restrict__ mkey, float* __restrict__ den,
                           int E, int H) {
  int gid = blockIdx.x * blockDim.x + threadIdx.x;
  if (gid >= E) return;
  int s = src[gid], d = dst[gid];
  for (int h = 0; h < H; ++h) {
    float e = lrelu(el[s * H + h] + er[d * H + h]);
    atomicAdd(&den[d * H + h], __expf(e - fdec(mkey[d * H + h])));
  }
}

// Pass 3: out[dst] += alpha * z[src]; 64 threads per edge (f = feature channel).
// Scalar operands (src,dst,el,er,m,den) are wave-uniform; z reads & atomics coalesced.
__global__ __launch_bounds__(256)
void edge_agg(const int* __restrict__ src, const int* __restrict__ dst,
              const float* __restrict__ el, const float* __restrict__ er,
              const unsigned* __restrict__ mkey, const float* __restrict__ den,
              const float* __restrict__ z, float* __restrict__ out,
              long long E, int H) {
  long long gid = (long long)blockIdx.x * blockDim.x + threadIdx.x;
  if (gid >= E * 64) return;
  long long e = gid >> 6;
  int f = (int)(gid & 63);
  int h = (H == 2) ? (f >> 5) : 0;
  int s = src[e], d = dst[e];
  float ec    = lrelu(el[s * H + h] + er[d * H + h]);
  float alpha = __expf(ec - fdec(mkey[d * H + h])) / den[d * H + h];
  atomicAdd(&out[(long long)d * 64 + f], alpha * z[(long long)s * 64 + f]);
}

// h = elu(agg + b1)   (in place)
__global__ void bias_elu(float* __restrict__ hbuf, const float* __restrict__ b, long long n64) {
  long long gid = (long long)blockIdx.x * blockDim.x + threadIdx.x;
  if (gid >= n64) return;
  float v = hbuf[gid] + b[gid & 63];
  hbuf[gid] = v > 0.f ? v : __expf(v) - 1.f;
}

// out += b2
__global__ void bias_add(float* __restrict__ out, const float* __restrict__ b, long long n64) {
  long long gid = (long long)blockIdx.x * blockDim.x + threadIdx.x;
  if (gid >= n64) return;
  out[gid] += b[gid & 63];
}

// ---------------------------------------------------------------------------
extern "C" void kernel_launch(void* const* d_in, const int* in_sizes, int n_in,
                              void* d_out, int out_size, void* d_ws, size_t ws_size,
                              hipStream_t stream) {
  const float* x   = (const float*)d_in[0];
  const int*   src = (const int*)  d_in[1];
  const int*   dst = (const int*)  d_in[2];
  const float* W1  = (const float*)d_in[3];
  const float* al1 = (const float*)d_in[4];
  const float* ar1 = (const float*)d_in[5];
  const float* b1  = (const float*)d_in[6];
  const float* W2  = (const float*)d_in[7];
  const float* al2 = (const float*)d_in[8];
  const float* ar2 = (const float*)d_in[9];
  const float* b2  = (const float*)d_in[10];

  const int IN_F = 128;
  const int N = in_sizes[0] / IN_F;           // 50000
  const long long E = in_sizes[1];            // 850000
  float* out = (float*)d_out;

  // ---- workspace layout (zero-init buffers contiguous) -------------------
  float* ws = (float*)d_ws;
  float*    z1    = ws;             ws += (long long)N * 64;
  float*    hbuf  = ws;             ws += (long long)N * 64;   // zeroed (agg1 -> h)
  unsigned* mkey1 = (unsigned*)ws;  ws += (long long)N * 2;    // zeroed (key 0 == -inf)
  float*    den1  = ws;             ws += (long long)N * 2;    // zeroed
  unsigned* mkey2 = (unsigned*)ws;  ws += N;                   // zeroed
  float*    den2  = ws;             ws += N;                   // zeroed
  float*    z2    = ws;             ws += (long long)N * 64;
  float*    el1   = ws;             ws += (long long)N * 2;
  float*    er1   = ws;             ws += (long long)N * 2;
  float*    el2   = ws;             ws += N;
  float*    er2   = ws;             ws += N;
  (void)ws_size; (void)n_in; (void)out_size;

  const long long zcount = (long long)N * 64 + (long long)N * 6; // hbuf..den2
  fill_zero<<<1024, 256, 0, stream>>>(hbuf, zcount);
  fill_zero<<<1024, 256, 0, stream>>>(out, (long long)N * 64);

  // ---- layer 1 ------------------------------------------------------------
  const int rowBlocks = (N + 15) / 16;                 // 3125
  dim3 gemmGrid((rowBlocks + 7) / 8);                  // 8 waves / block
  gemm_f32_wmma<<<gemmGrid, 256, 0, stream>>>(x, W1, z1, N, IN_F);

  attn_coef_h2<<<(N * 2 + 255) / 256, 256, 0, stream>>>(z1, al1, ar1, el1, er1, N);
  edge_max  <<<(int)((E + 255) / 256), 256, 0, stream>>>(src, dst, el1, er1, mkey1, (int)E, 2);
  edge_denom<<<(int)((E + 255) / 256), 256, 0, stream>>>(src, dst, el1, er1, mkey1, den1, (int)E, 2);
  edge_agg  <<<(int)((E * 64 + 255) / 256), 256, 0, stream>>>(src, dst, el1, er1, mkey1, den1,
                                                              z1, hbuf, E, 2);
  bias_elu<<<(int)(((long long)N * 64 + 255) / 256), 256, 0, stream>>>(hbuf, b1, (long long)N * 64);

  // ---- layer 2 ------------------------------------------------------------
  gemm_f32_wmma<<<gemmGrid, 256, 0, stream>>>(hbuf, W2, z2, N, 64);

  attn_coef_h1<<<(N + 255) / 256, 256, 0, stream>>>(z2, al2, ar2, el2, er2, N);
  edge_max  <<<(int)((E + 255) / 256), 256, 0, stream>>>(src, dst, el2, er2, mkey2, (int)E, 1);
  edge_denom<<<(int)((E + 255) / 256), 256, 0, stream>>>(src, dst, el2, er2, mkey2, den2, (int)E, 1);
  edge_agg  <<<(int)((E * 64 + 255) / 256), 256, 0, stream>>>(src, dst, el2, er2, mkey2, den2,
                                                              z2, out, E, 1);
  bias_add<<<(int)(((long long)N * 64 + 255) / 256), 256, 0, stream>>>(out, b2, (long long)N * 64);
}